// ConditionalRandomField_8667244003737
// MI455X (gfx1250) — compile-verified
//
#include <hip/hip_runtime.h>
#include <math.h>

// Problem sizes from the reference setup_inputs(): B=1024, T=512, K=50.
#define B_  1024
#define T_  512
#define K_  50
#define KP  64          // K padded to multiple of 16 for WMMA
#define PSTR 68         // padded LDS row stride (floats): 4*l mod 64 distinct -> conflict-free

typedef float v2f __attribute__((ext_vector_type(2)));
typedef float v8f __attribute__((ext_vector_type(8)));

#define NEG_BIG  (-1.0e30f)
#define NEG_BIG2 (-2.0e30f)   // pre-clamp sentinel for padding-column emissions
#define LOG2E    1.44269504088896340736f
#define LN2      0.69314718055994530942f

// Raw transcendental ops: v_exp_f32 (exp2) / v_log_f32 (log2), ~1ulp, no fixup code.
__device__ __forceinline__ float fast_exp(float x) {
    return __builtin_amdgcn_exp2f(x * LOG2E);
}
__device__ __forceinline__ float fast_log(float x) {
    return __builtin_amdgcn_logf(x) * LN2;
}

// 16-lane max butterfly: one fused v_max_num_f32_dpp per stage (4 VALU total).
// dst = max(dpp(src), src); after 4 stages every lane of each 16-group holds the max.
__device__ __forceinline__ float rowmax16(float x) {
    float a, b;
    asm("v_max_num_f32_dpp %0, %1, %1 quad_perm:[1,0,3,2] row_mask:0xf bank_mask:0xf bound_ctrl:1"
        : "=v"(a) : "v"(x));                     // lane ^ 1
    asm("v_max_num_f32_dpp %0, %1, %1 quad_perm:[2,3,0,1] row_mask:0xf bank_mask:0xf bound_ctrl:1"
        : "=v"(b) : "v"(a));                     // lane ^ 2
    asm("v_max_num_f32_dpp %0, %1, %1 row_half_mirror row_mask:0xf bank_mask:0xf bound_ctrl:1"
        : "=v"(a) : "v"(b));                     // lane ^ 7  (crosses quads)
    asm("v_max_num_f32_dpp %0, %1, %1 row_mirror row_mask:0xf bank_mask:0xf bound_ctrl:1"
        : "=v"(b) : "v"(a));                     // lane ^ 15 (crosses 8-groups)
    return b;
}

// ---------------------------------------------------------------------------
// Precompute expT[i][j] = exp(transitions[i][j]) padded with zeros to 64x64.
// ---------------------------------------------------------------------------
__global__ void crf_expT_kernel(const float* __restrict__ trans,
                                float* __restrict__ expT) {
    int idx = blockIdx.x * blockDim.x + threadIdx.x;
    if (idx >= KP * KP) return;
    int i = idx >> 6, j = idx & 63;
    float v = 0.0f;
    if (i < K_ && j < K_) v = fast_exp(trans[i * K_ + j]);
    expT[idx] = v;
}

// ---------------------------------------------------------------------------
// Forward recursion (log-denominator). One block = 16 batch rows, 4 waves.
// Wave w owns columns [16w, 16w+16) of alpha. Per step:
//   P = exp(alpha - rowmax)  (LDS, row-major, stride PSTR)
//   C = P @ expT             (16 x v_wmma_f32_16x16x4_f32, dual accumulators)
//   alpha' = log(C) + rowmax + emit   (masked update)
// ---------------------------------------------------------------------------
__launch_bounds__(128, 1)
__global__ void crf_forward_kernel(const float* __restrict__ logits,
                                   const float* __restrict__ expT,
                                   const float* __restrict__ startT,
                                   const float* __restrict__ endT,
                                   const int*   __restrict__ mask,
                                   float* __restrict__ log_den) {
    __shared__ __align__(16) float Pbuf[16 * PSTR];   // P (and final scores), row-major
    __shared__ __align__(16) float part[16 * 4];      // partial row maxima: [row][wave]

    const int tid  = threadIdx.x;
    const int w    = tid >> 5;        // wave id 0..3 = column tile
    const int lane = tid & 31;
    const int half = lane >> 4;       // 0 or 1
    const int l16  = lane & 15;
    const int col  = w * 16 + l16;    // global column 0..63
    const int ccol = (col < K_) ? col : (K_ - 1);  // clamped column for loads
    const bool colok = (col < K_);
    const int b0   = blockIdx.x * 16; // first batch row of this block

    // Loop-invariant B-fragments of expT for this wave's column tile.
    // B-matrix 4x16 f32: VGPR0 = rows {K0,K2}, VGPR1 = rows {K1,K3} across halves.
    v2f efrag[16];
#pragma unroll
    for (int kb = 0; kb < 16; ++kb) {
        const int r0 = 4 * kb + 2 * half;
        v2f e;
        e.x = expT[r0 * KP + col];
        e.y = expT[(r0 + 1) * KP + col];
        efrag[kb] = e;
    }

    float alpha[8];   // this wave's 16x16 alpha tile (C layout: row = v + 8*half)
    float mrow[8];    // rowmax used to build current P

    // t = 0: alpha0 = start + logits[:,0,:]
    const float startc = startT[ccol];
#pragma unroll
    for (int v = 0; v < 8; ++v) {
        const int row = v + 8 * half;
        const int b   = b0 + row;
        const float a = startc + logits[((size_t)b * T_) * K_ + ccol];
        alpha[v] = colok ? a : NEG_BIG;
    }
    // rowmax: tile-local via DPP butterfly, cross-wave via LDS [row][wave]
    {
        float r[8];
#pragma unroll
        for (int v = 0; v < 8; ++v) r[v] = rowmax16(alpha[v]);
        if (l16 == 0) {     // single EXEC toggle for all 8 partial stores
#pragma unroll
            for (int v = 0; v < 8; ++v) part[(v + 8 * half) * 4 + w] = r[v];
        }
    }
    __syncthreads();
#pragma unroll
    for (int v = 0; v < 8; ++v) {
        const int row = v + 8 * half;
        const float4 pm = *(const float4*)&part[row * 4];
        const float m = fmaxf(fmaxf(pm.x, pm.y), fmaxf(pm.z, pm.w));
        mrow[v] = m;
        Pbuf[row * PSTR + col] = fast_exp(alpha[v] - m);
    }
    __syncthreads();

    for (int t = 1; t < T_; ++t) {
        // Hoisted global loads + off-chain prep: pre[v] = mrow[v] + emit'
        // (emit' = NEG_BIG2 on padding columns; the clamp below restores NEG_BIG).
        float pre[8];
        int   mk[8];
#pragma unroll
        for (int v = 0; v < 8; ++v) {
            const int row = v + 8 * half;
            const int b   = b0 + row;
            const float e = logits[((size_t)b * T_ + t) * K_ + ccol];
            pre[v] = mrow[v] + (colok ? e : NEG_BIG2);
            mk[v]  = mask[b * T_ + t];
        }

        // C = P @ expT for this wave's column tile, K=64 in 16 chunks of 4.
        // Two independent accumulator chains to halve the WMMA dependency depth.
        v8f c0 = {0.f, 0.f, 0.f, 0.f, 0.f, 0.f, 0.f, 0.f};
        v8f c1 = {0.f, 0.f, 0.f, 0.f, 0.f, 0.f, 0.f, 0.f};
#pragma unroll
        for (int kb = 0; kb < 8; ++kb) {
            // A-matrix 16x4 f32: lane M = l16; VGPR0/1 = {K0,K1} (half0), {K2,K3} (half1)
            const v2f a0 = *(const v2f*)&Pbuf[l16 * PSTR + 4 * (2 * kb)     + 2 * half];
            const v2f a1 = *(const v2f*)&Pbuf[l16 * PSTR + 4 * (2 * kb + 1) + 2 * half];
            c0 = __builtin_amdgcn_wmma_f32_16x16x4_f32(
                     false, a0, false, efrag[2 * kb],     (short)0, c0, false, false);
            c1 = __builtin_amdgcn_wmma_f32_16x16x4_f32(
                     false, a1, false, efrag[2 * kb + 1], (short)0, c1, false, false);
        }

        // alpha' = log(C) + pre, masked; then rowmax of new alpha (DPP butterfly)
        float r[8];
#pragma unroll
        for (int v = 0; v < 8; ++v) {
            // chain: add, v_log, v_fmac(ln2,·,pre), v_max, v_cndmask
            float cand = fast_log(c0[v] + c1[v]) + pre[v];
            cand = fmaxf(cand, NEG_BIG);      // clamps log(0) and padding columns
            const float anew = mk[v] ? cand : alpha[v];
            alpha[v] = anew;
            r[v] = rowmax16(anew);
        }
        if (l16 == 0) {     // single EXEC toggle for all 8 partial stores
#pragma unroll
            for (int v = 0; v < 8; ++v) part[(v + 8 * half) * 4 + w] = r[v];
        }
        __syncthreads();   // partials visible; all waves done reading old P
#pragma unroll
        for (int v = 0; v < 8; ++v) {
            const int row = v + 8 * half;
            const float4 pm = *(const float4*)&part[row * 4];
            const float m = fmaxf(fmaxf(pm.x, pm.y), fmaxf(pm.z, pm.w));
            mrow[v] = m;
            Pbuf[row * PSTR + col] = fast_exp(alpha[v] - m);
        }
        __syncthreads();   // new P visible for next step
    }

    // Final: logsumexp(alpha + end_transitions) per row.
    const float endc = endT[ccol];
#pragma unroll
    for (int v = 0; v < 8; ++v) {
        const int row = v + 8 * half;
        const float x = alpha[v] + endc;
        Pbuf[row * PSTR + col] = colok ? x : NEG_BIG;
    }
    __syncthreads();
    if (tid < 16) {
        const int row = tid;
        float m = NEG_BIG;
        for (int cc = 0; cc < KP; ++cc) m = fmaxf(m, Pbuf[row * PSTR + cc]);
        float s = 0.0f;
        for (int cc = 0; cc < KP; ++cc) s += fast_exp(Pbuf[row * PSTR + cc] - m);
        log_den[b0 + row] = m + fast_log(s);
    }
}

// ---------------------------------------------------------------------------
// Log-numerator: one wave32 per batch row, lanes stride over T.
// ---------------------------------------------------------------------------
__global__ void crf_numerator_kernel(const float* __restrict__ logits,
                                     const float* __restrict__ trans,
                                     const float* __restrict__ startT,
                                     const float* __restrict__ endT,
                                     const int*   __restrict__ tags,
                                     const int*   __restrict__ mask,
                                     float* __restrict__ log_num) {
    const int wid  = (blockIdx.x * blockDim.x + threadIdx.x) >> 5;
    const int lane = threadIdx.x & 31;
    if (wid >= B_) return;
    const int*   tg = tags + (size_t)wid * T_;
    const int*   mk = mask + (size_t)wid * T_;
    const float* lg = logits + (size_t)wid * T_ * K_;

    float acc  = 0.0f;
    int   msum = 0;
    for (int t = lane; t < T_; t += 32) {
        const int m = mk[t];
        msum += m;
        const float fm = (float)m;
        if (t >= 1)      acc += trans[tg[t - 1] * K_ + tg[t]] * fm;  // transition score
        if (t <= T_ - 2) acc += lg[(size_t)t * K_ + tg[t]] * fm;     // emission score
    }
#pragma unroll
    for (int s = 16; s >= 1; s >>= 1) {
        acc  += __shfl_xor(acc, s);
        msum += __shfl_xor(msum, s);
    }
    if (lane == 0) {
        const int last = msum - 1;
        const int lt   = tg[last];
        float score = startT[tg[0]] + acc + endT[lt]
                    + lg[(size_t)(T_ - 1) * K_ + lt] * (float)mk[T_ - 1];
        log_num[wid] = score;
    }
}

// ---------------------------------------------------------------------------
// Deterministic final reduction: out = sum_b (num[b] - den[b]).
// ---------------------------------------------------------------------------
__global__ void crf_reduce_kernel(const float* __restrict__ num,
                                  const float* __restrict__ den,
                                  float* __restrict__ out) {
    __shared__ float sm[256];
    const int tid = threadIdx.x;
    float s = 0.0f;
    for (int i = tid; i < B_; i += 256) s += num[i] - den[i];
    sm[tid] = s;
    __syncthreads();
    for (int stride = 128; stride >= 1; stride >>= 1) {
        if (tid < stride) sm[tid] += sm[tid + stride];
        __syncthreads();
    }
    if (tid == 0) out[0] = sm[0];
}

// ---------------------------------------------------------------------------
extern "C" void kernel_launch(void* const* d_in, const int* in_sizes, int n_in,
                              void* d_out, int out_size, void* d_ws, size_t ws_size,
                              hipStream_t stream) {
    (void)in_sizes; (void)n_in; (void)out_size; (void)ws_size;
    const float* logits = (const float*)d_in[0];
    const float* trans  = (const float*)d_in[1];
    const float* startT = (const float*)d_in[2];
    const float* endT   = (const float*)d_in[3];
    const int*   tags   = (const int*)d_in[4];
    const int*   mask   = (const int*)d_in[5];

    float* ws   = (float*)d_ws;
    float* expT = ws;                  // KP*KP = 4096 floats
    float* den  = ws + KP * KP;        // B_ floats
    float* num  = den + B_;            // B_ floats

    crf_expT_kernel<<<(KP * KP + 255) / 256, 256, 0, stream>>>(trans, expT);
    crf_forward_kernel<<<B_ / 16, 128, 0, stream>>>(logits, expT, startT, endT,
                                                    mask, den);
    crf_numerator_kernel<<<B_ / 8, 256, 0, stream>>>(logits, trans, startT, endT,
                                                     tags, mask, num);
    crf_reduce_kernel<<<1, 256, 0, stream>>>(num, den, (float*)d_out);
}